// ModelTop2GateWithoutOneHot_69741678952706
// MI455X (gfx1250) — compile-verified
//
#include <hip/hip_runtime.h>
#include <hip/hip_bf16.h>

// Shapes fixed by the reference
#define S_TOK 8192
#define N_EXP 64
#define CAP   256
#define EC    (N_EXP * CAP)          // 16384 floats per token per output tensor
#define EPS_F32 1.1920929e-07f

typedef float v2f __attribute__((ext_vector_type(2)));
typedef float v4f __attribute__((ext_vector_type(4)));
typedef float v8f __attribute__((ext_vector_type(8)));

// -------------------------------------------------------------------------
// Kernel 1: per-token gate parameters.
// One block per token, 64 threads (one per expert).
// Produces: staged gates[S,64], normalized g1n/g2n, flattened nonzero
// positions p1/p2 (= e*CAP + c, or -1 if dropped).
// -------------------------------------------------------------------------
__global__ void gate_params_kernel(const float* __restrict__ logits,
                                   const float* __restrict__ mask1,
                                   const float* __restrict__ mask2,
                                   const float* __restrict__ loc1,
                                   const float* __restrict__ loc2,
                                   float* __restrict__ ws_gates,
                                   float* __restrict__ g1n_out,
                                   float* __restrict__ g2n_out,
                                   int*   __restrict__ p1_out,
                                   int*   __restrict__ p2_out)
{
    const int s = blockIdx.x;
    const int t = threadIdx.x;              // expert index 0..63

    __shared__ float red[N_EXP];
    __shared__ float shg[N_EXP];
    __shared__ int   ired[N_EXP];
    __shared__ int   se1, se2, sc1, sc2;

    // ---- softmax over 64 logits (max-subtract, like jax.nn.softmax) ----
    const float lg = logits[s * N_EXP + t];
    red[t] = lg;
    __syncthreads();
    for (int o = 32; o > 0; o >>= 1) {
        if (t < o) red[t] = fmaxf(red[t], red[t + o]);
        __syncthreads();
    }
    const float mx = red[0];
    __syncthreads();
    const float ex = expf(lg - mx);
    red[t] = ex;
    __syncthreads();
    for (int o = 32; o > 0; o >>= 1) {
        if (t < o) red[t] += red[t + o];
        __syncthreads();
    }
    const float gate = ex / red[0];
    shg[t] = gate;
    ws_gates[s * N_EXP + t] = gate;

    // ---- e1 / e2 from one-hot masks (max-index reduction) ----
    const float m1 = mask1[s * N_EXP + t];
    const float m2 = mask2[s * N_EXP + t];
    __syncthreads();
    ired[t] = (m1 > 0.5f) ? t : -1;
    __syncthreads();
    for (int o = 32; o > 0; o >>= 1) {
        if (t < o) ired[t] = max(ired[t], ired[t + o]);
        __syncthreads();
    }
    if (t == 0) se1 = ired[0];
    __syncthreads();
    ired[t] = (m2 > 0.5f) ? t : -1;
    __syncthreads();
    for (int o = 32; o > 0; o >>= 1) {
        if (t < o) ired[t] = max(ired[t], ired[t + o]);
        __syncthreads();
    }
    if (t == 0) se2 = ired[0];
    __syncthreads();

    // ---- c1 / c2 from one-hot location rows (each thread scans 4 slots) ----
    int c = -1;
    #pragma unroll
    for (int k = 0; k < 4; ++k) {
        const int i = t + N_EXP * k;
        if (loc1[(size_t)s * CAP + i] > 0.5f) c = i;
    }
    ired[t] = c;
    __syncthreads();
    for (int o = 32; o > 0; o >>= 1) {
        if (t < o) ired[t] = max(ired[t], ired[t + o]);
        __syncthreads();
    }
    if (t == 0) sc1 = ired[0];
    __syncthreads();
    c = -1;
    #pragma unroll
    for (int k = 0; k < 4; ++k) {
        const int i = t + N_EXP * k;
        if (loc2[(size_t)s * CAP + i] > 0.5f) c = i;
    }
    ired[t] = c;
    __syncthreads();
    for (int o = 32; o > 0; o >>= 1) {
        if (t < o) ired[t] = max(ired[t], ired[t + o]);
        __syncthreads();
    }
    if (t == 0) sc2 = ired[0];
    __syncthreads();

    if (t == 0) {
        const int e1 = se1, e2 = se2, c1 = sc1, c2 = sc2;
        const float g1 = (e1 >= 0) ? shg[e1] : 0.0f;   // == einsum(gates, mask1)
        const float g2 = (e2 >= 0) ? shg[e2] : 0.0f;
        const float den = fmaxf(g1 + g2, EPS_F32);     // clip(a_min=eps)
        g1n_out[s] = g1 / den;
        g2n_out[s] = g2 / den;
        p1_out[s] = (e1 >= 0 && c1 >= 0) ? (e1 * CAP + c1) : -1;
        p2_out[s] = (e2 >= 0 && c2 >= 0) ? (e2 * CAP + c2) : -1;
    }
}

// -------------------------------------------------------------------------
// Kernel 2: column sums of gates[S,64] and mask1[S,64] via
// V_WMMA_F32_16X16X4_F32 with an all-ones A matrix: D = 1*B + C accumulates
// 4 rows worth of column sums per instruction. A is all-ones, so the exact
// K-slot permutation of B's VGPR layout is irrelevant; only N = lane%16
// matters (matches C/D layout). 64 blocks x 4 waves; block b covers rows
// [b*128, b*128+128), wave w covers expert columns [16w, 16w+16).
// Writes per-block partials (no float atomics -> deterministic).
// -------------------------------------------------------------------------
__global__ void colsum_wmma_kernel(const float* __restrict__ gates,
                                   const float* __restrict__ mask1,
                                   float* __restrict__ partial_g,
                                   float* __restrict__ partial_m)
{
    const int b    = blockIdx.x;
    const int wave = threadIdx.x >> 5;           // 0..3
    const int lane = threadIdx.x & 31;
    const int col  = wave * 16 + (lane & 15);    // expert column for this lane
    const int rsel = (lane < 16) ? 0 : 2;        // which pair of rows this half covers
    const int rbase = b * 128;

    v2f a; a.x = 1.0f; a.y = 1.0f;               // all-ones A (16x4)
    v8f cg = {}; v8f cm = {};

    for (int r = 0; r < 128; r += 4) {
        const int row = rbase + r + rsel;
        v2f bg, bm;
        bg.x = gates[(size_t)row       * N_EXP + col];
        bg.y = gates[(size_t)(row + 1) * N_EXP + col];
        bm.x = mask1[(size_t)row       * N_EXP + col];
        bm.y = mask1[(size_t)(row + 1) * N_EXP + col];
        cg = __builtin_amdgcn_wmma_f32_16x16x4_f32(false, a, false, bg,
                                                   (short)0, cg, false, false);
        cm = __builtin_amdgcn_wmma_f32_16x16x4_f32(false, a, false, bm,
                                                   (short)0, cm, false, false);
    }

    // Every row of D is identical (= column sums); lanes 16-31 duplicate N.
    if (lane < 16) {
        partial_g[b * N_EXP + col] = cg[0];
        partial_m[b * N_EXP + col] = cm[0];
    }
}

// -------------------------------------------------------------------------
// Kernel 3: l_aux = mean_e(me*ce) * E^2 = E * sum_e(me*ce). Deterministic
// serial reduction of the 64 block-partials per expert, then LDS tree.
// -------------------------------------------------------------------------
__global__ void laux_kernel(const float* __restrict__ partial_g,
                            const float* __restrict__ partial_m,
                            float* __restrict__ out)
{
    __shared__ float red[N_EXP];
    const int e = threadIdx.x;                   // 64 threads
    float sg = 0.0f, sm = 0.0f;
    for (int b = 0; b < 64; ++b) {
        sg += partial_g[b * N_EXP + e];
        sm += partial_m[b * N_EXP + e];
    }
    const float me = sg / (float)S_TOK;
    const float ce = sm / (float)S_TOK;
    red[e] = me * ce;
    __syncthreads();
    for (int o = 32; o > 0; o >>= 1) {
        if (e < o) red[e] += red[e + o];
        __syncthreads();
    }
    if (e == 0) out[0] = red[0] * (float)N_EXP;
}

// -------------------------------------------------------------------------
// Kernel 4 (bandwidth-critical): dense fill of combine_weights and
// dispatch_mask. One block per token; each token owns 16384 floats in each
// tensor with <=2 nonzeros at p1/p2. Base index is 1 + s*16384 (only 4B
// aligned), so: 3 scalar head stores, 4095 aligned 16B stores, 1 scalar
// tail. Writes 1.07 GB total with coalesced global_store_b128.
// Output (1.07 GB) >> L2 (192 MB) and is written exactly once -> use
// non-temporal store hints (ISA §7.3 TH=NT) so the stream does not thrash
// L2-resident inputs/workspace. Native ext-vector v4f used so the
// nontemporal builtin accepts the pointer type.
// -------------------------------------------------------------------------
__device__ __forceinline__ float pick2(int i, int p1, int p2, float g1, float g2) {
    return (i == p1) ? g1 : ((i == p2) ? g2 : 0.0f);
}

__global__ void __launch_bounds__(256)
fill_kernel(const float* __restrict__ g1n, const float* __restrict__ g2n,
            const int* __restrict__ p1a, const int* __restrict__ p2a,
            float* __restrict__ out)
{
    const int s   = blockIdx.x;
    const int tid = threadIdx.x;

    const float g1 = g1n[s];
    const float g2 = g2n[s];
    const int   p1 = p1a[s];
    const int   p2 = p2a[s];
    const float d1 = (g1 != 0.0f) ? 1.0f : 0.0f;   // dispatch = bool(combine)
    const float d2 = (g2 != 0.0f) ? 1.0f : 0.0f;

    float* cw = out + 1 + (size_t)s * EC;                          // combine row
    float* dm = out + 1 + (size_t)S_TOK * EC + (size_t)s * EC;     // dispatch row

    // head (i = 0,1,2) and tail (i = 16383): scalar stores
    if (tid < 4) {
        const int i = (tid < 3) ? tid : (EC - 1);
        __builtin_nontemporal_store(pick2(i, p1, p2, g1, g2), &cw[i]);
        __builtin_nontemporal_store(pick2(i, p1, p2, d1, d2), &dm[i]);
    }

    // body: 4095 16B chunks, aligned (index 1 + s*16384 + 3 ≡ 0 mod 4)
    v4f* __restrict__ cw4 = (v4f*)(cw + 3);
    v4f* __restrict__ dm4 = (v4f*)(dm + 3);
    for (int j = tid; j < (EC - 4) / 4; j += 256) {   // 4095 chunks
        const int i0 = 3 + 4 * j;
        v4f v, m;
        v.x = pick2(i0 + 0, p1, p2, g1, g2);
        v.y = pick2(i0 + 1, p1, p2, g1, g2);
        v.z = pick2(i0 + 2, p1, p2, g1, g2);
        v.w = pick2(i0 + 3, p1, p2, g1, g2);
        m.x = pick2(i0 + 0, p1, p2, d1, d2);
        m.y = pick2(i0 + 1, p1, p2, d1, d2);
        m.z = pick2(i0 + 2, p1, p2, d1, d2);
        m.w = pick2(i0 + 3, p1, p2, d1, d2);
        __builtin_nontemporal_store(v, &cw4[j]);
        __builtin_nontemporal_store(m, &dm4[j]);
    }
}

// -------------------------------------------------------------------------
extern "C" void kernel_launch(void* const* d_in, const int* in_sizes, int n_in,
                              void* d_out, int out_size, void* d_ws, size_t ws_size,
                              hipStream_t stream) {
    const float* logits = (const float*)d_in[0];   // [S,E]
    const float* mask1  = (const float*)d_in[1];   // [S,E]
    const float* mask2  = (const float*)d_in[2];   // [S,E]
    const float* loc1   = (const float*)d_in[3];   // [S,C]
    const float* loc2   = (const float*)d_in[4];   // [S,C]
    float* out = (float*)d_out;                    // [1 + S*E*C + S*E*C]

    // Workspace layout (bytes): ~2.2 MB total
    char* ws = (char*)d_ws;
    float* g1n      = (float*)(ws + 0);                         // S floats
    float* g2n      = (float*)(ws + 32768);                     // S floats
    int*   p1       = (int*)  (ws + 65536);                     // S ints
    int*   p2       = (int*)  (ws + 98304);                     // S ints
    float* gates    = (float*)(ws + 131072);                    // S*E floats (2 MB)
    float* pg       = (float*)(ws + 131072 + 2097152);          // 64*64 floats
    float* pm       = pg + 64 * N_EXP;                          // 64*64 floats

    gate_params_kernel<<<S_TOK, N_EXP, 0, stream>>>(logits, mask1, mask2, loc1, loc2,
                                                    gates, g1n, g2n, p1, p2);
    colsum_wmma_kernel<<<64, 128, 0, stream>>>(gates, mask1, pg, pm);
    laux_kernel<<<1, N_EXP, 0, stream>>>(pg, pm, out);
    fill_kernel<<<S_TOK, 256, 0, stream>>>(g1n, g2n, p1, p2, out);
}